// DistillationLoss_79267916415457
// MI455X (gfx1250) — compile-verified
//
#include <hip/hip_runtime.h>
#include <stdint.h>

// Problem constants (match reference)
#define NB      8192
#define NK      64
#define NGLOB   16384
#define INV_T   0.5f          // 1/TEMPERATURE

// LSE kernel tiling
#define ROWS_PB 16            // rows per block
#define CW      256           // chunk width (columns)
#define NCH     (NB / CW)     // 32 chunks
#define RS      (CW + 4)      // padded LDS row stride (floats) -> bank-conflict free
#define THREADS 256
#define WAVES   (THREADS / 32)
#define LDPT    ((ROWS_PB * CW / 4) / THREADS)   // float4 loads per thread per chunk (=4)

typedef float v2f __attribute__((ext_vector_type(2)));
typedef float v8f __attribute__((ext_vector_type(8)));

#if defined(__gfx1250__)
  #define IS_GFX1250 1
#else
  #define IS_GFX1250 0
#endif

#if IS_GFX1250 && __has_builtin(__builtin_amdgcn_wmma_f32_16x16x4_f32)
  #define USE_WMMA 1
#else
  #define USE_WMMA 0
#endif

#define USE_ASYNC IS_GFX1250

// ---------------------------------------------------------------------------
// Kernel 1: per-row logsumexp of logits/T over a 16-row stripe.
// One HBM read per element: async-copy 16x256 chunks into LDS (double
// buffered), online max with per-chunk accumulator rescale, exp-sum reduced
// on the matrix pipe via V_WMMA_F32_16X16X4_F32 with B = ones.
// ---------------------------------------------------------------------------
__global__ __launch_bounds__(THREADS) void k_lse(const float* __restrict__ logits,
                                                 float* __restrict__ Lout) {
  __shared__ float buf[2][ROWS_PB * RS];
  __shared__ float rowmax[ROWS_PB];
  __shared__ float Mrow[ROWS_PB];
  __shared__ float Frow[ROWS_PB];
  __shared__ float wsum[WAVES][ROWS_PB];

  const int t    = threadIdx.x;
  const int lane = t & 31;
  const int w    = t >> 5;
  const int rb   = blockIdx.x * ROWS_PB;
  const float* gbase = logits + (size_t)rb * NB;

  if (t < ROWS_PB) Mrow[t] = -3.0e38f;

#if USE_WMMA
  v8f acc = {};
#else
  float accs = 0.0f;
#endif
  __syncthreads();

  auto issue_chunk = [&](int c, int bi) {
#pragma unroll
    for (int i = 0; i < LDPT; ++i) {
      const int e4  = i * THREADS + t;        // float4 index in 16xCW tile
      const int row = e4 / (CW / 4);
      const int c4  = e4 % (CW / 4);
      const int eoff = row * NB + c * CW + c4 * 4;   // element offset from gbase
#if USE_ASYNC
      const unsigned loff = (unsigned)(uintptr_t)&buf[bi][row * RS + c4 * 4];
      const unsigned voff = (unsigned)(eoff * 4);    // byte offset
      asm volatile("global_load_async_to_lds_b128 %0, %1, %2"
                   :: "v"(loff), "v"(voff), "s"(gbase) : "memory");
#else
      *(float4*)&buf[bi][row * RS + c4 * 4] = *(const float4*)(gbase + eoff);
#endif
    }
  };

  issue_chunk(0, 0);
  for (int c = 0; c < NCH; ++c) {
    const int bi = c & 1;
    if (c + 1 < NCH) issue_chunk(c + 1, (c + 1) & 1);
#if USE_ASYNC
    if (c + 1 < NCH) asm volatile("s_wait_asynccnt 4" ::: "memory");
    else             asm volatile("s_wait_asynccnt 0" ::: "memory");
#endif
    __syncthreads();   // chunk c resident in LDS for all waves

    // --- per-row chunk max (16 threads per row, half-wave shuffle reduce) ---
    {
      const int mr = t >> 4;
      const int ml = t & 15;
      const float* rp = &buf[bi][mr * RS];
      float m = -3.0e38f;
#pragma unroll
      for (int i = 0; i < CW / 16; ++i) m = fmaxf(m, rp[ml + i * 16]);
#pragma unroll
      for (int s = 8; s > 0; s >>= 1) m = fmaxf(m, __shfl_xor(m, s, 32));
      if (ml == 0) rowmax[mr] = m;
    }
    __syncthreads();
    if (t < ROWS_PB) {
      const float mo = Mrow[t];
      const float mn = fmaxf(mo, rowmax[t]);
      Frow[t] = __expf((mo - mn) * INV_T);   // accumulator rescale factor
      Mrow[t] = mn;
    }
    __syncthreads();

    const float mrow_l  = Mrow[lane & 15];
    const int   colbase = w * (CW / WAVES);        // 32 disjoint cols per wave
#if USE_WMMA
    // rescale accumulator: C/D layout -> VGPR j holds row j (lanes 0-15) / j+8
#pragma unroll
    for (int j = 0; j < 8; ++j) acc[j] *= Frow[j + ((lane < 16) ? 0 : 8)];
#pragma unroll
    for (int s = 0; s < (CW / WAVES) / 4; ++s) {
      // A (16x4 f32): lanes 0-15 -> K={0,1}, lanes 16-31 -> K={2,3}, M=lane%16
      const int col = colbase + s * 4 + ((lane < 16) ? 0 : 2);
      const float2 x = *(const float2*)&buf[bi][(lane & 15) * RS + col];
      v2f a;
      a.x = __expf((x.x - mrow_l) * INV_T);
      a.y = __expf((x.y - mrow_l) * INV_T);
      v2f ones; ones.x = 1.0f; ones.y = 1.0f;
      // D[m,n] += sum_k A[m,k] : row-sum on the matrix pipe
      acc = __builtin_amdgcn_wmma_f32_16x16x4_f32(false, a, false, ones,
                                                  (short)0, acc, false, false);
    }
#else
    accs *= Frow[lane & 15];
    for (int s = 0; s < (CW / WAVES) / 4; ++s) {
      const int col = colbase + s * 4 + ((lane < 16) ? 0 : 2);
      const float2 x = *(const float2*)&buf[bi][(lane & 15) * RS + col];
      accs += __expf((x.x - mrow_l) * INV_T) + __expf((x.y - mrow_l) * INV_T);
    }
#endif
    __syncthreads();   // all waves done with buf[bi]; safe to refill next iter
  }

  // --- extract per-wave row sums, combine across 8 waves ---
#if USE_WMMA
  if (lane == 0) {
#pragma unroll
    for (int j = 0; j < 8; ++j) wsum[w][j] = acc[j];        // rows 0..7  (N=0)
  }
  if (lane == 16) {
#pragma unroll
    for (int j = 0; j < 8; ++j) wsum[w][8 + j] = acc[j];    // rows 8..15 (N=0)
  }
#else
  accs += __shfl_down(accs, 16, 32);
  if (lane < 16) wsum[w][lane] = accs;
#endif
  __syncthreads();
  if (t < ROWS_PB) {
    float S = 0.0f;
#pragma unroll
    for (int ww = 0; ww < WAVES; ++ww) S += wsum[ww][t];
    Lout[rb + t] = Mrow[t] * INV_T + logf(S);   // logsumexp of logits/T
  }
}

// ---------------------------------------------------------------------------
// g2l table: global index -> local row (-1 if absent)
// ---------------------------------------------------------------------------
__global__ __launch_bounds__(256) void k_fill_g2l(int* __restrict__ g2l) {
  const int i = blockIdx.x * 256 + threadIdx.x;
  if (i < NGLOB) g2l[i] = -1;
}
__global__ __launch_bounds__(256) void k_scatter_g2l(const int* __restrict__ bidx,
                                                     int* __restrict__ g2l) {
  const int i = blockIdx.x * 256 + threadIdx.x;
  if (i < NB) {
    int g = bidx[i];
    if (g >= 0 && g < NGLOB) g2l[g] = i;
  }
}

// ---------------------------------------------------------------------------
// Kernel 2: per-row sparse KL term. 64 threads (2 waves) per row.
// Scatter-set semantics: last duplicate wins; diag forced to 1.0.
// ---------------------------------------------------------------------------
__global__ __launch_bounds__(NK) void k_rowloss(const float* __restrict__ logits,
                                                const float* __restrict__ scores,
                                                const int*   __restrict__ tidx,
                                                const int*   __restrict__ g2l,
                                                const float* __restrict__ L,
                                                float* __restrict__ rowloss) {
  const int r = blockIdx.x;
  const int k = threadIdx.x;
  __shared__ int   colsS[NK];
  __shared__ float wred[2];

  int gidx = tidx[r * NK + k];
  gidx = gidx < 0 ? 0 : (gidx > NGLOB - 1 ? NGLOB - 1 : gidx);
  const int   lpos = g2l[gidx];
  const float sc   = scores[r * NK + k];
  const bool valid = (lpos >= 0) && (lpos != r);   // diag overwritten by 1.0
  colsS[k] = valid ? lpos : (-1 - k);              // unique sentinel if invalid
  __syncthreads();

  bool survive = valid;
  const int myc = colsS[k];
  for (int j = k + 1; j < NK; ++j) survive = survive && (colsS[j] != myc);
  const float tval = (survive && sc > 0.0f) ? sc : 0.0f;

  // row sum (fixed-tree, deterministic)
  float rp = tval;
#pragma unroll
  for (int s = 16; s > 0; s >>= 1) rp += __shfl_xor(rp, s, 32);
  if ((k & 31) == 0) wred[k >> 5] = rp;
  __syncthreads();
  const float rs = 1.0f + wred[0] + wred[1];       // >= 1, clip(1e-8) moot
  const float Lr = L[r];

  float contrib = 0.0f;
  if (tval > 0.0f) {
    const float tp  = tval / rs;
    const float lgp = logits[(size_t)r * NB + myc] * INV_T - Lr;
    contrib = tp * (logf(tp) - lgp);
  }
  if (k == 0) {                                    // diagonal entry t = 1/rs
    const float tp  = 1.0f / rs;
    const float lgp = logits[(size_t)r * NB + r] * INV_T - Lr;
    contrib += tp * (logf(tp) - lgp);
  }
  float cp = contrib;
#pragma unroll
  for (int s = 16; s > 0; s >>= 1) cp += __shfl_xor(cp, s, 32);
  __syncthreads();                                  // all reads of wred done
  if ((k & 31) == 0) wred[k >> 5] = cp;
  __syncthreads();
  if (k == 0) rowloss[r] = wred[0] + wred[1];
}

// ---------------------------------------------------------------------------
// Kernel 3: deterministic final reduction, * T^2 / B
// ---------------------------------------------------------------------------
__global__ __launch_bounds__(256) void k_final(const float* __restrict__ rowloss,
                                               float* __restrict__ out) {
  __shared__ float sm[256];
  const int t = threadIdx.x;
  float a = 0.0f;
  for (int i = t; i < NB; i += 256) a += rowloss[i];
  sm[t] = a;
  __syncthreads();
  for (int s = 128; s > 0; s >>= 1) {
    if (t < s) sm[t] += sm[t + s];
    __syncthreads();
  }
  if (t == 0) out[0] = sm[0] * (4.0f / (float)NB);   // * T^2 / B
}

// ---------------------------------------------------------------------------
extern "C" void kernel_launch(void* const* d_in, const int* in_sizes, int n_in,
                              void* d_out, int out_size, void* d_ws, size_t ws_size,
                              hipStream_t stream) {
  const float* logits = (const float*)d_in[0];   // [B,B] f32
  const float* scores = (const float*)d_in[1];   // [B,K] f32
  const int*   bidx   = (const int*)d_in[2];     // [B]   i32
  const int*   tind   = (const int*)d_in[3];     // [B,K] i32

  float* Lws     = (float*)d_ws;                 // [B] logsumexp
  float* rowloss = Lws + NB;                     // [B] per-row loss
  int*   g2l     = (int*)(rowloss + NB);         // [NGLOB] map

  k_fill_g2l   <<<(NGLOB + 255) / 256, 256, 0, stream>>>(g2l);
  k_scatter_g2l<<<(NB    + 255) / 256, 256, 0, stream>>>(bidx, g2l);
  k_lse        <<<NB / ROWS_PB, THREADS, 0, stream>>>(logits, Lws);
  k_rowloss    <<<NB, NK, 0, stream>>>(logits, scores, tind, g2l, Lws, rowloss);
  k_final      <<<1, 256, 0, stream>>>(rowloss, (float*)d_out);
}